// attention_4071628997298
// MI455X (gfx1250) — compile-verified
//
#include <hip/hip_runtime.h>
#include <math.h>

typedef __attribute__((ext_vector_type(2))) float v2f;
typedef __attribute__((ext_vector_type(4))) float f4;
typedef __attribute__((ext_vector_type(8))) float v8f;

#define B_DIM 32
#define T_DIM 2048
#define D_DIM 1024
#define CHUNKS 16
#define ROWS_PER_CHUNK (T_DIM / CHUNKS) /* 128 */
#define ATTN_WAVES 8
#define ATTN_TPB (ATTN_WAVES * 32)

// ---------------------------------------------------------------------------
// Kernel 1: s_fi[m][n] = sum_k s[m][k] * W_s[k][n] + b_s[n]   (M=32,N=1024,K=1024)
// One wave per 16x16 C tile, V_WMMA_F32_16X16X4_F32, K-step 4.
// ---------------------------------------------------------------------------
__global__ __launch_bounds__(32)
void proj_s_kernel(const float* __restrict__ s, const float* __restrict__ Ws,
                   const float* __restrict__ bs, float* __restrict__ s_fi) {
  const int lane = threadIdx.x;
  const int half = lane >> 4;       // 0: lanes 0-15, 1: lanes 16-31
  const int mn   = lane & 15;       // A row / B col within tile
  const int mt   = blockIdx.y * 16; // M tile base
  const int nt   = blockIdx.x * 16; // N tile base
  const int kv   = half * 2;        // K sub-offset held by this half-wave

  v8f acc = {};
  for (int k = 0; k < D_DIM; k += 4) {
    v2f a, b;
    const float* ap = s + (mt + mn) * D_DIM + (k + kv);
    a.x = ap[0];                              // A[m][k+kv]
    a.y = ap[1];                              // A[m][k+kv+1]
    b.x = Ws[(size_t)(k + kv) * D_DIM + nt + mn];       // B[k+kv][n]
    b.y = Ws[(size_t)(k + kv + 1) * D_DIM + nt + mn];   // B[k+kv+1][n]
    acc = __builtin_amdgcn_wmma_f32_16x16x4_f32(false, a, false, b,
                                                (short)0, acc, false, false);
  }
  const float bias = bs[nt + mn];
#pragma unroll
  for (int j = 0; j < 8; ++j) {
    const int row = j + half * 8;             // C layout: VGPR j -> M=j (+8 upper lanes)
    s_fi[(size_t)(mt + row) * D_DIM + nt + mn] = acc[j] + bias;
  }
}

// ---------------------------------------------------------------------------
// Kernel 2: v[m][n] = sum_k s_fi[m][k] * W_h[n][k]    (B matrix = W_h^T)
// ---------------------------------------------------------------------------
__global__ __launch_bounds__(32)
void proj_v_kernel(const float* __restrict__ s_fi, const float* __restrict__ Wh,
                   float* __restrict__ v) {
  const int lane = threadIdx.x;
  const int half = lane >> 4;
  const int mn   = lane & 15;
  const int mt   = blockIdx.y * 16;
  const int nt   = blockIdx.x * 16;
  const int kv   = half * 2;

  v8f acc = {};
  for (int k = 0; k < D_DIM; k += 4) {
    v2f a, b;
    const float* ap = s_fi + (size_t)(mt + mn) * D_DIM + (k + kv);
    a.x = ap[0];
    a.y = ap[1];
    const float* bp = Wh + (size_t)(nt + mn) * D_DIM + (k + kv); // W_h[n][k] = B[k][n]
    b.x = bp[0];
    b.y = bp[1];
    acc = __builtin_amdgcn_wmma_f32_16x16x4_f32(false, a, false, b,
                                                (short)0, acc, false, false);
  }
#pragma unroll
  for (int j = 0; j < 8; ++j) {
    const int row = j + half * 8;
    v[(size_t)(mt + row) * D_DIM + nt + mn] = acc[j];
  }
}

// ---------------------------------------------------------------------------
// Kernel 3: fused scores + online softmax + context accumulation.
// grid = (CHUNKS, B). Each block streams 128 rows of h[b] exactly once.
// e[t] = h[b,t] . v[b]   (b_h dropped: softmax-invariant constant)
// Per-wave online softmax state; LDS merge across 8 waves; partial
// (m, l, unnormalized acc[1024]) written to workspace.
// ---------------------------------------------------------------------------
__global__ __launch_bounds__(ATTN_TPB)
void attn_pass_kernel(const float* __restrict__ h, const float* __restrict__ v,
                      float* __restrict__ pml, float* __restrict__ pacc) {
  const int chunk = blockIdx.x;
  const int b     = blockIdx.y;
  const int lane  = threadIdx.x & 31;
  const int w     = threadIdx.x >> 5;

  const float* hb = h + (size_t)b * T_DIM * D_DIM;
  const float* vb = v + (size_t)b * D_DIM;

  // v[b] resident in VGPRs: lane covers d = 128*j + lane*4 + {0..3}
  f4 vreg[8];
#pragma unroll
  for (int j = 0; j < 8; ++j)
    vreg[j] = *(const f4*)(vb + j * 128 + lane * 4);

  float m = -INFINITY;
  float l = 0.0f;
  f4 acc[8];
#pragma unroll
  for (int j = 0; j < 8; ++j) acc[j] = (f4)0.0f;

  const int t0 = chunk * ROWS_PER_CHUNK;
  for (int i = 0; i < ROWS_PER_CHUNK / ATTN_WAVES; ++i) {
    const int t = t0 + w + i * ATTN_WAVES;
    const f4* hr = (const f4*)(hb + (size_t)t * D_DIM);
    f4 hreg[8];
    float dot = 0.0f;
#pragma unroll
    for (int j = 0; j < 8; ++j) {
      hreg[j] = __builtin_nontemporal_load(&hr[j * 32 + lane]); // streaming: NT hint
      dot += hreg[j].x * vreg[j].x + hreg[j].y * vreg[j].y +
             hreg[j].z * vreg[j].z + hreg[j].w * vreg[j].w;
    }
#pragma unroll
    for (int off = 16; off >= 1; off >>= 1)
      dot += __shfl_xor(dot, off, 32);

    const float mn    = fmaxf(m, dot);
    const float scale = __expf(m - mn);    // first iter: exp(-inf)=0
    const float p     = __expf(dot - mn);
    l = l * scale + p;
#pragma unroll
    for (int j = 0; j < 8; ++j)
      acc[j] = acc[j] * scale + p * hreg[j];
    m = mn;
  }

  // merge 8 waves within the block via LDS
  __shared__ float sm[ATTN_WAVES];
  __shared__ float sl[ATTN_WAVES];
  __shared__ __align__(16) float sacc[ATTN_WAVES * D_DIM]; // 32 KB of 320 KB
  if (lane == 0) { sm[w] = m; sl[w] = l; }
  __syncthreads();

  float M = -INFINITY;
#pragma unroll
  for (int w2 = 0; w2 < ATTN_WAVES; ++w2) M = fmaxf(M, sm[w2]);
  float L = 0.0f;
#pragma unroll
  for (int w2 = 0; w2 < ATTN_WAVES; ++w2) L += sl[w2] * __expf(sm[w2] - M);

  const float myscale = __expf(m - M);
#pragma unroll
  for (int j = 0; j < 8; ++j)
    *(f4*)&sacc[w * D_DIM + j * 128 + lane * 4] = acc[j] * myscale;
  __syncthreads();

  const int d = threadIdx.x * 4; // 256 threads * 4 = 1024
  f4 sum = (f4)0.0f;
#pragma unroll
  for (int w2 = 0; w2 < ATTN_WAVES; ++w2)
    sum += *(const f4*)&sacc[w2 * D_DIM + d];

  const size_t idx = (size_t)b * CHUNKS + chunk;
  if (threadIdx.x == 0) { pml[idx * 2 + 0] = M; pml[idx * 2 + 1] = L; }
  *(f4*)&pacc[idx * D_DIM + d] = sum;   // unnormalized, referenced to M
}

// ---------------------------------------------------------------------------
// Kernel 4: merge the 16 chunk-partials per batch, normalize, write out.
// ---------------------------------------------------------------------------
__global__ __launch_bounds__(256)
void merge_kernel(const float* __restrict__ pml, const float* __restrict__ pacc,
                  float* __restrict__ out) {
  const int b = blockIdx.x;
  float M = -INFINITY;
#pragma unroll
  for (int c = 0; c < CHUNKS; ++c)
    M = fmaxf(M, pml[((size_t)b * CHUNKS + c) * 2 + 0]);
  float L = 0.0f;
#pragma unroll
  for (int c = 0; c < CHUNKS; ++c)
    L += pml[((size_t)b * CHUNKS + c) * 2 + 1] *
         __expf(pml[((size_t)b * CHUNKS + c) * 2 + 0] - M);

  const int d = threadIdx.x * 4;
  f4 sum = (f4)0.0f;
#pragma unroll
  for (int c = 0; c < CHUNKS; ++c) {
    const float wgt = __expf(pml[((size_t)b * CHUNKS + c) * 2 + 0] - M);
    sum += wgt * *(const f4*)&pacc[((size_t)b * CHUNKS + c) * D_DIM + d];
  }
  const float invL = 1.0f / L;
  *(f4*)&out[(size_t)b * D_DIM + d] = sum * invL;
}

// ---------------------------------------------------------------------------
extern "C" void kernel_launch(void* const* d_in, const int* in_sizes, int n_in,
                              void* d_out, int out_size, void* d_ws, size_t ws_size,
                              hipStream_t stream) {
  const float* s   = (const float*)d_in[0]; // [32,1,1024]
  const float* h   = (const float*)d_in[1]; // [32,2048,1024]
  const float* W_s = (const float*)d_in[2]; // [1024,1024]
  const float* b_s = (const float*)d_in[3]; // [1024]
  const float* W_h = (const float*)d_in[4]; // [1024,1024]
  // d_in[5] = b_h: softmax-invariant per-batch constant -> provably unused.
  float* out = (float*)d_out;               // [32,1024]

  float* ws   = (float*)d_ws;
  float* s_fi = ws;                               // 32*1024
  float* v    = ws + B_DIM * D_DIM;               // 32*1024
  float* pml  = ws + 2 * B_DIM * D_DIM;           // 512*2
  float* pacc = ws + 2 * B_DIM * D_DIM + 2 * B_DIM * CHUNKS; // 512*1024

  dim3 gproj(D_DIM / 16, B_DIM / 16);             // (64, 2)
  proj_s_kernel<<<gproj, 32, 0, stream>>>(s, W_s, b_s, s_fi);
  proj_v_kernel<<<gproj, 32, 0, stream>>>(s_fi, W_h, v);

  dim3 gattn(CHUNKS, B_DIM);                      // 512 blocks
  attn_pass_kernel<<<gattn, ATTN_TPB, 0, stream>>>(h, v, pml, pacc);

  merge_kernel<<<B_DIM, 256, 0, stream>>>(pml, pacc, out);
}